// Attention_79224966742124
// MI455X (gfx1250) — compile-verified
//
#include <hip/hip_runtime.h>
#include <hip/hip_bf16.h>
#include <math.h>
#include <stdint.h>

typedef __attribute__((ext_vector_type(2))) float v2f;
typedef __attribute__((ext_vector_type(8))) float v8f;

#define B_  2
#define H_  16
#define S_  2048
#define D_  64
#define QT  16                       // query rows per block
#define NTHREADS 512
#define WAVES (NTHREADS / 32)        // 16 waves
#define KPW (S_ / WAVES)             // 128 keys per wave
#define KT  (KPW / 16)               // 8 key-tiles of 16 per wave

__global__ __launch_bounds__(NTHREADS) void attn_fused_kernel(
    const float* __restrict__ Q, const float* __restrict__ K,
    const float* __restrict__ V, const uint8_t* __restrict__ MSK,
    const float* __restrict__ ADJ, float* __restrict__ OUT,
    float* __restrict__ P)
{
    __shared__ float pt[WAVES][16][17];   // per-wave P-tile transpose staging
    __shared__ float wstat[WAVES][16];    // per-wave row stats
    __shared__ float red[16];             // reduced row stat (max, then 1/sum)
    __shared__ float outred[QT][D_];      // cross-wave output reduction

    const int tid  = threadIdx.x;
    const int w    = tid >> 5;
    const int lane = tid & 31;
    const int n    = lane & 15;                 // N / M position within 16-group
    const int koff = (lane < 16) ? 0 : 2;       // K offset for A/B fragments
    const int half = (lane < 16) ? 0 : 8;       // M offset for C/D rows

    const int bid   = blockIdx.x;
    const int bh    = bid >> 7;                 // / (S_/QT)
    const int qtile = bid & 127;
    const int b     = bh >> 4;
    const int qbase = qtile << 4;

    const float*   qhead = Q   + ((size_t)bh * S_ + qbase) * D_;
    const float*   khead = K   + (size_t)bh * S_ * D_;
    const float*   vhead = V   + (size_t)bh * S_ * D_;
    const float*   adj0  = ADJ + ((size_t)bh * S_ + qbase) * S_;
    const uint8_t* msk0  = MSK + ((size_t)b  * S_ + qbase) * S_;
    float*         prow0 = P   + ((size_t)bh * S_ + qbase) * S_;

    // ---------------- Q A-fragments (16x64 -> 16 frags of 16x4) ----------------
    v2f aq[16];
    {
        const float* qrow = qhead + (size_t)n * D_ + koff;
        #pragma unroll
        for (int i = 0; i < 16; ++i)
            aq[i] = *(const v2f*)(qrow + 4 * i);
    }

    // ---------------- QK^T : scores for 16 rows x 128 keys in registers -------
    v8f acc[KT];
    #pragma unroll
    for (int kt = 0; kt < KT; ++kt) {
        const int keybase = w * KPW + kt * 16;
        const float* krow = khead + (size_t)(keybase + n) * D_ + koff;
        v8f c = {0.f, 0.f, 0.f, 0.f, 0.f, 0.f, 0.f, 0.f};
        #pragma unroll
        for (int i = 0; i < 16; ++i) {
            v2f bk = *(const v2f*)(krow + 4 * i);
            c = __builtin_amdgcn_wmma_f32_16x16x4_f32(
                    false, aq[i], false, bk, (short)0, c, false, false);
        }
        acc[kt] = c;
    }

    // ---------------- bias + scale + mask -------------------------------------
    const float scale = 0.125f; // 1/sqrt(64)
    #pragma unroll
    for (int kt = 0; kt < KT; ++kt) {
        const int keybase = w * KPW + kt * 16;
        #pragma unroll
        for (int v = 0; v < 8; ++v) {
            const int m = v + half;
            const size_t off = (size_t)m * S_ + keybase + n;
            const float s = (acc[kt][v] + adj0[off]) * scale;
            acc[kt][v] = (msk0[off] != 0) ? -__builtin_inff() : s;
        }
    }

    // ---------------- softmax: row max across 2048 (wave chunk -> LDS) --------
    {
        #pragma unroll
        for (int v = 0; v < 8; ++v) {
            float mx = acc[0][v];
            #pragma unroll
            for (int kt = 1; kt < KT; ++kt) mx = fmaxf(mx, acc[kt][v]);
            #pragma unroll
            for (int o = 1; o < 16; o <<= 1)
                mx = fmaxf(mx, __shfl_xor(mx, o, 32));
            if (n == 0) wstat[w][v + half] = mx;
        }
    }
    __syncthreads();
    if (tid < 16) {
        float mx = wstat[0][tid];
        #pragma unroll
        for (int ww = 1; ww < WAVES; ++ww) mx = fmaxf(mx, wstat[ww][tid]);
        red[tid] = mx;
    }
    __syncthreads();

    // ---------------- exp + row sum -------------------------------------------
    {
        #pragma unroll
        for (int v = 0; v < 8; ++v) {
            const float rmx = red[v + half];
            float s = 0.f;
            #pragma unroll
            for (int kt = 0; kt < KT; ++kt) {
                const float e = __expf(acc[kt][v] - rmx);
                acc[kt][v] = e;
                s += e;
            }
            #pragma unroll
            for (int o = 1; o < 16; o <<= 1) s += __shfl_xor(s, o, 32);
            if (n == 0) wstat[w][v + half] = s;   // safe: phase-1 reads done
        }
    }
    __syncthreads();
    if (tid < 16) {
        float s = 0.f;
        #pragma unroll
        for (int ww = 0; ww < WAVES; ++ww) s += wstat[ww][tid];
        red[tid] = 1.0f / s;                      // all old reads of red are done
    }
    __syncthreads();

    // ---------------- normalize, stream p_attn to global ----------------------
    #pragma unroll
    for (int kt = 0; kt < KT; ++kt) {
        const int keybase = w * KPW + kt * 16;
        #pragma unroll
        for (int v = 0; v < 8; ++v) {
            const int m = v + half;
            const float p = acc[kt][v] * red[m];
            acc[kt][v] = p;
            prow0[(size_t)m * S_ + keybase + n] = p;
        }
    }

    // ---------------- PV: out[16x64] += P[16x128] x V[128x64] per wave --------
    v8f oacc[4];
    #pragma unroll
    for (int t = 0; t < 4; ++t)
        oacc[t] = (v8f){0.f, 0.f, 0.f, 0.f, 0.f, 0.f, 0.f, 0.f};

    #pragma unroll
    for (int kt = 0; kt < KT; ++kt) {
        const int keybase = w * KPW + kt * 16;
        // stage C-layout P tile -> LDS for A-layout reload (transpose-in-lane)
        #pragma unroll
        for (int v = 0; v < 8; ++v)
            pt[w][v + half][n] = acc[kt][v];
        __syncthreads();   // uniform across all waves; orders stage vs. read

        #pragma unroll
        for (int f = 0; f < 4; ++f) {
            v2f a;
            a.x = pt[w][n][4 * f + koff];
            a.y = pt[w][n][4 * f + koff + 1];
            const float* vrow = vhead + (size_t)(keybase + 4 * f + koff) * D_;
            #pragma unroll
            for (int t = 0; t < 4; ++t) {
                v2f bv;
                bv.x = vrow[t * 16 + n];
                bv.y = vrow[D_ + t * 16 + n];
                oacc[t] = __builtin_amdgcn_wmma_f32_16x16x4_f32(
                              false, a, false, bv, (short)0, oacc[t], false, false);
            }
        }
        __syncthreads();
    }

    // ---------------- cross-wave reduction of output --------------------------
    for (int i = tid; i < QT * D_; i += NTHREADS)
        (&outred[0][0])[i] = 0.f;
    __syncthreads();
    #pragma unroll
    for (int t = 0; t < 4; ++t)
        #pragma unroll
        for (int v = 0; v < 8; ++v)
            atomicAdd(&outred[v + half][t * 16 + n], oacc[t][v]);
    __syncthreads();

    float* og = OUT + ((size_t)bh * S_ + qbase) * D_;
    for (int i = tid; i < QT * D_; i += NTHREADS)
        og[i] = (&outred[0][0])[i];
}

extern "C" void kernel_launch(void* const* d_in, const int* in_sizes, int n_in,
                              void* d_out, int out_size, void* d_ws, size_t ws_size,
                              hipStream_t stream) {
    (void)in_sizes; (void)n_in; (void)d_ws; (void)ws_size; (void)out_size;
    const float*   Q   = (const float*)d_in[0];
    const float*   K   = (const float*)d_in[1];
    const float*   V   = (const float*)d_in[2];
    const uint8_t* M   = (const uint8_t*)d_in[3];
    const float*   ADJ = (const float*)d_in[4];
    float* OUT = (float*)d_out;
    float* P   = OUT + (size_t)B_ * H_ * S_ * D_;   // tuple: (out, p_attn) concatenated

    dim3 grid(B_ * H_ * (S_ / QT));   // 4096 blocks
    attn_fused_kernel<<<grid, NTHREADS, 0, stream>>>(Q, K, V, M, ADJ, OUT, P);
}